// YOLOV2Loss_58179626991827
// MI455X (gfx1250) — compile-verified
//
#include <hip/hip_runtime.h>
#include <hip/hip_bf16.h>
#include <math.h>

// ---- problem constants (match reference) ----
#define ANUM 5
#define CLSN 80
#define GSZ  19
#define HWN  361         // 19*19
#define NN   128
#define KCH  85          // 5 + CLS
#define CCH  425         // A * KCH
#define IMGF 608.0f
#define IOUTH 0.6f
#define NUNITS (NN * HWN)          // 46208
#define BLK 256

typedef __attribute__((ext_vector_type(2))) float v2f;
typedef __attribute__((ext_vector_type(8))) float v8f;

__device__ __constant__ float c_aw[ANUM] = {0.57273f, 1.87446f, 3.33843f, 7.88282f, 9.77052f};
__device__ __constant__ float c_ah[ANUM] = {0.677385f, 2.06253f, 5.47434f, 3.52778f, 9.16828f};

__device__ __forceinline__ float sigm(float x) { return 1.0f / (1.0f + __expf(-x)); }

__global__ void yolo_ws_init(float* ws) {
    if (threadIdx.x < 8) ws[threadIdx.x] = 0.0f;
}

__global__ void __launch_bounds__(BLK)
yolo_loss_partial(const float* __restrict__ outp,
                  const float* __restrict__ gtp,
                  float* __restrict__ ws)
{
    const int tid = threadIdx.x;
    const int u   = blockIdx.x * BLK + tid;

    // per-thread partial sums (accumulators 0..6; 7 = padding)
    float S1 = 0.f, S2 = 0.f, S3 = 0.f, S4 = 0.f, S5 = 0.f, Cp = 0.f, Cn = 0.f;

    if (u < NUNITS) {
        const int n  = u / HWN;
        const int hw = u - n * HWN;
        const int h  = hw / GSZ;
        const int w  = hw - h * GSZ;
        const float gxf = (float)w, gyf = (float)h;   // grid = (x=w, y=h)
        const float invG = 1.0f / (float)GSZ;

        float iouv[ANUM], confv[ANUM], gconfv[ANUM];
        int   amax = 0;
        float imax = -3.4e38f;

        for (int a = 0; a < ANUM; ++a) {
            const int b = ((n * CCH + a * KCH) * HWN) + hw;
            // out: tx,ty,tw,th,tconf ; gt: xyxy(pixels), gconf
            const float tx = outp[b];
            const float ty = outp[b + 1 * HWN];
            const float tw = outp[b + 2 * HWN];
            const float th = outp[b + 3 * HWN];
            const float tc = outp[b + 4 * HWN];
            const float g0 = gtp[b];
            const float g1 = gtp[b + 1 * HWN];
            const float g2 = gtp[b + 2 * HWN];
            const float g3 = gtp[b + 3 * HWN];
            const float gc = gtp[b + 4 * HWN];

            const float cx = (sigm(tx) + gxf) * invG;
            const float cy = (sigm(ty) + gyf) * invG;
            const float pw = __expf(tw) * (c_aw[a] * invG);
            const float ph = __expf(th) * (c_ah[a] * invG);
            const float x0 = cx - 0.5f * pw, y0 = cy - 0.5f * ph;
            const float x1 = cx + 0.5f * pw, y1 = cy + 0.5f * ph;

            // wh-only IoU (mirrors reference: widths from xyxy diffs)
            const float w0 = x1 - x0, h0 = y1 - y0;
            const float gx0 = g0 / IMGF, gy0 = g1 / IMGF;
            const float gx1 = g2 / IMGF, gy1 = g3 / IMGF;
            const float w1 = gx1 - gx0, h1 = gy1 - gy0;
            const float s0 = w0 * h0, s1 = w1 * h1;
            const float inter = fminf(w0, w1) * fminf(h0, h1);
            const float iou = inter / (s0 + s1 - inter);

            // warm-up anchor-box term (unmasked mean): anchor xyxy at this cell
            const float ax0 = gxf * invG - 0.5f * c_aw[a] * invG;
            const float ay0 = gyf * invG - 0.5f * c_ah[a] * invG;
            const float ax1 = gxf * invG + 0.5f * c_aw[a] * invG;
            const float ay1 = gyf * invG + 0.5f * c_ah[a] * invG;
            const float d0 = x0 - ax0, d1 = y0 - ay0, d2 = x1 - ax1, d3 = y1 - ay1;
            S2 += d0 * d0 + d1 * d1 + d2 * d2 + d3 * d3;

            iouv[a]  = iou;
            confv[a] = sigm(tc);
            gconfv[a] = gc;
            if (iou > imax) { imax = iou; amax = a; }   // first-max, like jnp.argmax
        }

        // negative mask: g_conf<=0  OR  (g_conf>0 AND iou<TH)  ==  g<=0 || iou<TH
        for (int a = 0; a < ANUM; ++a) {
            if (gconfv[a] <= 0.f || iouv[a] < IOUTH) {
                S4 += confv[a] * confv[a];
                Cn += 1.f;
            }
        }

        // positive: argmax anchor with g_conf>0 (at most one per cell)
        if (gconfv[amax] > 0.f) {
            const int a = amax;
            const int b = ((n * CCH + a * KCH) * HWN) + hw;
            // recompute predicted xyxy for the winning anchor (cheap L0/L2 reload)
            const float cx = (sigm(outp[b]) + gxf) * invG;
            const float cy = (sigm(outp[b + 1 * HWN]) + gyf) * invG;
            const float pw = __expf(outp[b + 2 * HWN]) * (c_aw[a] * invG);
            const float ph = __expf(outp[b + 3 * HWN]) * (c_ah[a] * invG);
            const float x0 = cx - 0.5f * pw, y0 = cy - 0.5f * ph;
            const float x1 = cx + 0.5f * pw, y1 = cy + 0.5f * ph;
            const float e0 = x0 - gtp[b] / IMGF;
            const float e1 = y0 - gtp[b + 1 * HWN] / IMGF;
            const float e2 = x1 - gtp[b + 2 * HWN] / IMGF;
            const float e3 = y1 - gtp[b + 3 * HWN] / IMGF;
            S1 += e0 * e0 + e1 * e1 + e2 * e2 + e3 * e3;
            Cp = 1.f;
            const float ed = confv[a] - iouv[a];
            S3 += ed * ed;

            // class softmax + MSE, 3-pass streaming (no 80-reg blowup; L2-resident)
            const int cb = b + 5 * HWN;
            float mx = -3.4e38f;
            for (int c = 0; c < CLSN; ++c) mx = fmaxf(mx, outp[cb + c * HWN]);
            float se = 0.f;
            for (int c = 0; c < CLSN; ++c) se += __expf(outp[cb + c * HWN] - mx);
            const float inv = 1.0f / se;
            for (int c = 0; c < CLSN; ++c) {
                const float p = __expf(outp[cb + c * HWN] - mx) * inv;
                const float d = p - gtp[cb + c * HWN];
                S5 += d * d;
            }
        }
    }

    // ---- block reduction: 16x256 row-sum via chained V_WMMA_F32_16X16X4_F32 ----
    // LDS matrix: row j (accumulator) x col t (thread). Rows 8..15 zeroed.
    __shared__ float red[16 * BLK];
    red[0 * BLK + tid] = S1;
    red[1 * BLK + tid] = S2;
    red[2 * BLK + tid] = S3;
    red[3 * BLK + tid] = S4;
    red[4 * BLK + tid] = S5;
    red[5 * BLK + tid] = Cp;
    red[6 * BLK + tid] = Cn;
    red[7 * BLK + tid] = 0.f;
#pragma unroll
    for (int j = 8; j < 16; ++j) red[j * BLK + tid] = 0.f;
    __syncthreads();

    // All waves run the WMMA chain uniformly (EXEC must be all 1s for WMMA).
    // A-matrix 16x4 f32 layout: lanes 0-15 -> M=lane, {v0,v1}={K0,K1};
    //                           lanes 16-31 -> M=lane-16, {v0,v1}={K2,K3}.
    // B = all-ones 4x16  =>  D[m,n] += sum_k A[m,k]  (row sums, replicated over n).
    const int lane = tid & 31;
    const int m    = lane & 15;
    const int koff = (lane < 16) ? 0 : 2;
    v8f acc = {0.f, 0.f, 0.f, 0.f, 0.f, 0.f, 0.f, 0.f};
    v2f bone; bone.x = 1.0f; bone.y = 1.0f;
#pragma unroll 4
    for (int c = 0; c < 64; ++c) {
        v2f av;
        av.x = red[m * BLK + 4 * c + koff];
        av.y = red[m * BLK + 4 * c + koff + 1];
        acc = __builtin_amdgcn_wmma_f32_16x16x4_f32(
            false, av, false, bone, (short)0, acc, false, false);
    }

    // Lane 0 holds D[r,0] in acc[r] (C/D layout: VGPR r <-> M=r for lanes 0-15).
    if (tid == 0) {
#pragma unroll
        for (int r = 0; r < 7; ++r) atomicAdd(&ws[r], acc[r]);
    }
}

__global__ void yolo_finalize(const float* __restrict__ ws, float* __restrict__ outp) {
    const float S1 = ws[0], S2 = ws[1], S3 = ws[2], S4 = ws[3], S5 = ws[4];
    const float Cp = ws[5], Cn = ws[6];
    const float pos1 = S1 / fmaxf(4.0f * Cp, 1.0f);
    const float pos2 = S2 / (float)(NN * HWN * ANUM * 4);
    const float posL = pos1 + pos2;
    const float has  = S3 / fmaxf(Cp, 1.0f);
    const float noo  = S4 / fmaxf(Cn, 1.0f);
    const float cls  = S5 / fmaxf((float)CLSN * Cp, 1.0f);
    outp[0] = posL + has + noo + cls;
    outp[1] = posL;
    outp[2] = has;
    outp[3] = noo;
    outp[4] = cls;
}

extern "C" void kernel_launch(void* const* d_in, const int* in_sizes, int n_in,
                              void* d_out, int out_size, void* d_ws, size_t ws_size,
                              hipStream_t stream) {
    const float* out_t = (const float*)d_in[0];
    const float* gt_t  = (const float*)d_in[1];
    float* ws = (float*)d_ws;
    float* o  = (float*)d_out;

    yolo_ws_init<<<1, 32, 0, stream>>>(ws);
    const int blocks = (NUNITS + BLK - 1) / BLK;   // 181
    yolo_loss_partial<<<blocks, BLK, 0, stream>>>(out_t, gt_t, ws);
    yolo_finalize<<<1, 1, 0, stream>>>(ws, o);
}